// GlyphBag_90065464197716
// MI455X (gfx1250) — compile-verified
//
#include <hip/hip_runtime.h>
#include <math.h>

typedef __attribute__((ext_vector_type(2))) float v2f;
typedef __attribute__((ext_vector_type(8))) float v8f;

#define CHAR_DIM_  96
#define COLOR_DIM_ 16
#define GLYPH_DIM_ 1536
#define CHAR_EMB_  16
#define COLOR_EMB_ 4
#define GEMB_      20
#define HID_       32
#define MAXLEN_    64
#define HW_        1659   // 21*79
#define NWORDS_    48     // 1536/32

// ---------------------------------------------------------------------------
// Branch-free tanh: prefer gfx1250 hardware v_tanh_f32, else exp2/rcp TRANS
// sequence. Avoids libm tanhf's s_and_saveexec divergence inside the WMMA loop.
// ---------------------------------------------------------------------------
__device__ __forceinline__ float fast_tanh(float x) {
#if __has_builtin(__builtin_amdgcn_tanhf)
    return __builtin_amdgcn_tanhf(x);
#elif __has_builtin(__builtin_amdgcn_tanh_f32)
    return __builtin_amdgcn_tanh_f32(x);
#else
    // tanh(x) = 1 - 2/(exp2(2x*log2e) + 1); saturates correctly at +-inf
    float e = __builtin_amdgcn_exp2f(x * 2.885390081777927f);
    return 1.0f - 2.0f * __builtin_amdgcn_rcpf(e + 1.0f);
#endif
}

// ---------------------------------------------------------------------------
// Phase 1: per-sample bounded histogram -> sorted-unique bag -> embedding gather
// One workgroup per batch sample. LDS bitmap + ds_or atomics.
// ---------------------------------------------------------------------------
__global__ __launch_bounds__(256) void glyph_hist_kernel(
    const int* __restrict__ chars, const int* __restrict__ colors,
    const float* __restrict__ char_table, const float* __restrict__ col_table,
    float* __restrict__ emb_out, float* __restrict__ bag_out,
    int* __restrict__ lengths)
{
    __shared__ unsigned int bitmap[NWORDS_];
    __shared__ int idlist[MAXLEN_];

    const int b   = blockIdx.x;
    const int tid = threadIdx.x;

    if (tid < NWORDS_) bitmap[tid] = 0u;
    __syncthreads();

    const int base = b * HW_;
    for (int i = tid; i < HW_; i += 256) {
        if (i + 256 < HW_) {
            __builtin_prefetch(&chars[base + i + 256], 0, 0);
            __builtin_prefetch(&colors[base + i + 256], 0, 0);
        }
        int id = chars[base + i] * COLOR_DIM_ + colors[base + i];
        atomicOr(&bitmap[id >> 5], 1u << (id & 31));
    }
    __syncthreads();

    if (tid == 0) {
        int total = 0, cnt = 0;
        for (int w = 0; w < NWORDS_; ++w) {
            unsigned int m = bitmap[w];
            total += __popc(m);
            while (m && cnt < MAXLEN_) {
                int bit = __ffs(m) - 1;           // ascending -> sorted ids
                idlist[cnt++] = (w << 5) + bit;
                m &= m - 1;
            }
        }
        for (int i = cnt; i < MAXLEN_; ++i) idlist[i] = GLYPH_DIM_;
        lengths[b] = (total < MAXLEN_) ? total : MAXLEN_;
    }
    __syncthreads();

    if (tid < MAXLEN_) {
        int id   = idlist[tid];
        bool pad = (id >= GLYPH_DIM_);
        int ci = pad ? CHAR_DIM_  : (id >> 4);           // id / COLOR_DIM
        int co = pad ? COLOR_DIM_ : (id & (COLOR_DIM_ - 1));

        float* bp = bag_out + ((size_t)b * MAXLEN_ + tid) * 2;
        bp[0] = (float)ci;
        bp[1] = (float)co;

        // vectorized gather/scatter: 20-float rows are 16B-aligned (80 B),
        // char rows 64 B, col rows 16 B -> b128 traffic
        float4* ep = (float4*)(emb_out + ((size_t)b * MAXLEN_ + tid) * GEMB_);
        const float4* cr = (const float4*)(char_table + (size_t)ci * CHAR_EMB_);
        ep[0] = cr[0];
        ep[1] = cr[1];
        ep[2] = cr[2];
        ep[3] = cr[3];
        ep[4] = *(const float4*)(col_table + (size_t)co * COLOR_EMB_);
    }
}

// ---------------------------------------------------------------------------
// Phase 2: 64-step tanh RNN, 16 samples per wave via V_WMMA_F32_16X16X4_F32.
// D[16x32] = X[16x20]*Wih^T + H[16x32]*Whh^T + bias  (two 16-wide N blocks,
// K tiled by 4). H staged through per-wave LDS to convert D-layout -> A-layout.
// ---------------------------------------------------------------------------
__device__ __forceinline__ v8f wmma4(v2f a, v2f b, v8f c) {
    return __builtin_amdgcn_wmma_f32_16x16x4_f32(
        /*neg_a=*/false, a, /*neg_b=*/false, b,
        /*c_mod=*/(short)0, c, /*reuse_a=*/false, /*reuse_b=*/false);
}

__global__ __launch_bounds__(128) void glyph_rnn_kernel(
    const float* __restrict__ emb,  // [B,64,20]
    const float* __restrict__ Wih,  // [32,20]
    const float* __restrict__ Whh,  // [32,32]
    const float* __restrict__ bih,  // [32]
    const float* __restrict__ bhh,  // [32]
    const int*   __restrict__ lengths, // [B]
    float* __restrict__ h_out,      // [B,32]
    int B)
{
    __shared__ float lds_h[4][16 * HID_];   // per-wave H tile [m=16][n=32]

    const int lane = threadIdx.x & 31;
    const int wave = threadIdx.x >> 5;
    const int group = blockIdx.x * 4 + wave;     // 16 samples per wave
    const int sbase = group * 16;
    if (sbase >= B) return;                       // whole-wave uniform exit

    const int half = lane >> 4;                   // 0: K even pair / M 0..7 ; 1: odd
    const int mn   = lane & 15;                   // A: row m ; B/C/D: col n
    float* hbuf = lds_h[wave];

    // --- Load B-operands (weights transposed) into WMMA B layout ---
    // B element (k, n): vgpr = k&1, lane = (n&15) + 16*(k>>1 within chunk)
    v2f bw_ih[5][2];
#pragma unroll
    for (int kc = 0; kc < 5; ++kc)
#pragma unroll
        for (int nb = 0; nb < 2; ++nb) {
            int n = nb * 16 + mn;
            int k = kc * 4 + half * 2;
            bw_ih[kc][nb].x = Wih[n * GEMB_ + k];
            bw_ih[kc][nb].y = Wih[n * GEMB_ + k + 1];
        }
    v2f bw_hh[8][2];
#pragma unroll
    for (int kc = 0; kc < 8; ++kc)
#pragma unroll
        for (int nb = 0; nb < 2; ++nb) {
            int n = nb * 16 + mn;
            int k = kc * 4 + half * 2;
            bw_hh[kc][nb].x = Whh[n * HID_ + k];
            bw_hh[kc][nb].y = Whh[n * HID_ + k + 1];
        }

    const float bias0 = bih[mn] + bhh[mn];
    const float bias1 = bih[16 + mn] + bhh[16 + mn];

    // lengths for this lane's D rows: m = v + half*8
    float lenf[8];
#pragma unroll
    for (int v = 0; v < 8; ++v)
        lenf[v] = (float)lengths[sbase + half * 8 + v];

    // H state kept in D layout registers; init zero and mirror into LDS
    v8f hreg0, hreg1;
#pragma unroll
    for (int v = 0; v < 8; ++v) { hreg0[v] = 0.f; hreg1[v] = 0.f; }
#pragma unroll
    for (int v = 0; v < 8; ++v) {
        int m = v + half * 8;
        hbuf[m * HID_ + mn]      = 0.f;
        hbuf[m * HID_ + 16 + mn] = 0.f;
    }
    asm volatile("s_wait_dscnt 0" ::: "memory");

    for (int t = 0; t < MAXLEN_; ++t) {
        v8f acc0, acc1;
#pragma unroll
        for (int v = 0; v < 8; ++v) { acc0[v] = bias0; acc1[v] = bias1; }

        // --- X @ Wih^T : K = 20, five K-chunks of 4 ---
        const float* xrow =
            emb + ((size_t)(sbase + mn) * MAXLEN_ + t) * GEMB_;
#pragma unroll
        for (int kc = 0; kc < 5; ++kc) {
            v2f a;                                 // A(m=mn, k)
            a.x = xrow[kc * 4 + half * 2];
            a.y = xrow[kc * 4 + half * 2 + 1];
            acc0 = wmma4(a, bw_ih[kc][0], acc0);
            acc1 = wmma4(a, bw_ih[kc][1], acc1);
        }

        // --- H @ Whh^T : K = 32, eight K-chunks of 4, A from LDS ---
#pragma unroll
        for (int kc = 0; kc < 8; ++kc) {
            int k = kc * 4 + half * 2;
            v2f a;
            a.x = hbuf[mn * HID_ + k];
            a.y = hbuf[mn * HID_ + k + 1];
            acc0 = wmma4(a, bw_hh[kc][0], acc0);
            acc1 = wmma4(a, bw_hh[kc][1], acc1);
        }

        // --- tanh + length mask (D layout: row m = v + half*8) ---
        const float tf = (float)t;
#pragma unroll
        for (int v = 0; v < 8; ++v) {
            float n0 = fast_tanh(acc0[v]);
            float n1 = fast_tanh(acc1[v]);
            bool keep = tf < lenf[v];
            hreg0[v] = keep ? n0 : hreg0[v];
            hreg1[v] = keep ? n1 : hreg1[v];
        }

        // --- mirror H back to LDS for next step's A-layout reads ---
#pragma unroll
        for (int v = 0; v < 8; ++v) {
            int m = v + half * 8;
            hbuf[m * HID_ + mn]      = hreg0[v];
            hbuf[m * HID_ + 16 + mn] = hreg1[v];
        }
        asm volatile("s_wait_dscnt 0" ::: "memory");
    }

    // --- store final hidden state ---
#pragma unroll
    for (int v = 0; v < 8; ++v) {
        int m = v + half * 8;
        h_out[(size_t)(sbase + m) * HID_ + mn]      = hreg0[v];
        h_out[(size_t)(sbase + m) * HID_ + 16 + mn] = hreg1[v];
    }
}

// ---------------------------------------------------------------------------
extern "C" void kernel_launch(void* const* d_in, const int* in_sizes, int n_in,
                              void* d_out, int out_size, void* d_ws, size_t ws_size,
                              hipStream_t stream) {
    const int*   chars      = (const int*)  d_in[0];
    const int*   colors     = (const int*)  d_in[1];
    const float* char_table = (const float*)d_in[2];
    const float* col_table  = (const float*)d_in[3];
    const float* Wih        = (const float*)d_in[4];
    const float* Whh        = (const float*)d_in[5];
    const float* bih        = (const float*)d_in[6];
    const float* bhh        = (const float*)d_in[7];

    const int B = in_sizes[0] / HW_;   // 4096

    float* out   = (float*)d_out;
    float* h_out = out;                                        // [B,32]
    float* emb   = out + (size_t)B * HID_;                     // [B,64,20]
    float* bag   = emb + (size_t)B * MAXLEN_ * GEMB_;          // [B,64,2]
    int*   lens  = (int*)d_ws;                                 // [B]

    glyph_hist_kernel<<<B, 256, 0, stream>>>(
        chars, colors, char_table, col_table, emb, bag, lens);

    const int groups_per_block = 4;                 // 4 waves * 16 samples = 64
    int nblocks = (B + 16 * groups_per_block - 1) / (16 * groups_per_block);
    glyph_rnn_kernel<<<nblocks, 128, 0, stream>>>(
        emb, Wih, Whh, bih, bhh, lens, h_out, B);
}